// Model_36575941493205
// MI455X (gfx1250) — compile-verified
//
#include <hip/hip_runtime.h>

typedef __attribute__((ext_vector_type(2))) float v2f;
typedef __attribute__((ext_vector_type(8))) float v8f;

#define N_   100
#define T_   200
#define B_   128
#define P_   64
#define N2_  10000            // N*N
#define K_   20000            // T*N  GEMM contraction length
#define NPAD 112              // N padded to a multiple of 16
#define KS_  25               // K-split factor (parallelism)
#define KC_  800              // K per chunk (KS_*KC_ == K_)
#define TK_  32               // K rows staged in LDS per stage
#define NSTG (KC_ / TK_)      // 25 stages per chunk

// ---------------------------------------------------------------------------
// Kernel 1: F[r] = sum_k g[r,k] * weights[k]^2      (memory-bound SGEMV)
// 400 MB of g traffic -> the roofline term (~17 us @ 23.3 TB/s).
// ---------------------------------------------------------------------------
__global__ __launch_bounds__(256)
void f_gemv_kernel(const float* __restrict__ g,
                   const float* __restrict__ w,
                   float* __restrict__ F) {
  __shared__ float red[256];
  const int row = blockIdx.x;                 // 0..9999
  const int tid = threadIdx.x;                // 0..255
  const float4* grow = (const float4*)(g + (long)row * N2_);
  const float4* w4   = (const float4*)w;
  float acc = 0.f;
  for (int k4 = tid; k4 < N2_ / 4; k4 += 256) {
    if (k4 + 256 < N2_ / 4)
      __builtin_prefetch(&grow[k4 + 256], 0, 1);   // global_prefetch_b8
    const float4 gv = grow[k4];
    const float4 wv = w4[k4];
    acc += gv.x * (wv.x * wv.x) + gv.y * (wv.y * wv.y)
         + gv.z * (wv.z * wv.z) + gv.w * (wv.w * wv.w);
  }
  red[tid] = acc;
  __syncthreads();
  for (int s = 128; s > 0; s >>= 1) {
    if (tid < s) red[tid] += red[tid + s];
    __syncthreads();
  }
  if (tid == 0) F[row] = red[0];
}

// ---------------------------------------------------------------------------
// Kernel 2: row-softmax of W[t,n,m] = -alpha[t]^2 * F[n,m], stored TRANSPOSED
// and padded: fT[(t*N + m) * NPAD + n] = f[t,n,m]. One block per (t,n) row.
// ---------------------------------------------------------------------------
__global__ __launch_bounds__(128)
void softmax_kernel(const float* __restrict__ F,
                    const float* __restrict__ alphas,
                    float* __restrict__ fT) {
  __shared__ float red[128];
  const int bid = blockIdx.x;                 // t*N + n
  const int t   = bid / N_;
  const int n   = bid - t * N_;
  const int m   = threadIdx.x;                // 0..127 (100 active)
  const float a  = alphas[t];
  const float a2 = a * a;
  float w = -1e30f;
  if (m < N_) w = -a2 * F[n * N_ + m];
  red[m] = w;
  __syncthreads();
  for (int s = 64; s > 0; s >>= 1) {
    if (m < s) red[m] = fmaxf(red[m], red[m + s]);
    __syncthreads();
  }
  const float mx = red[0];
  __syncthreads();
  const float e = (m < N_) ? __expf(w - mx) : 0.f;
  red[m] = e;
  __syncthreads();
  for (int s = 64; s > 0; s >>= 1) {
    if (m < s) red[m] += red[m + s];
    __syncthreads();
  }
  const float inv = 1.f / red[0];
  if (m < N_) fT[(size_t)(t * N_ + m) * NPAD + n] = e * inv;
}

// ---------------------------------------------------------------------------
// Kernel 3: scatter-accumulate Xs[b, t, m] = sum_{p: x_i[b,p]==t} x[b,m,p].
// Thread m owns (b, *, m) exclusively -> plain RMW, no atomics. Xs pre-zeroed.
// ---------------------------------------------------------------------------
__global__ __launch_bounds__(128)
void xs_build_kernel(const float* __restrict__ x,
                     const int* __restrict__ xi,
                     float* __restrict__ Xs) {
  const int b = blockIdx.x;
  const int m = threadIdx.x;
  if (m >= N_) return;
  float*       xsb = Xs + (size_t)b * K_;
  const float* xb  = x + ((size_t)b * N_ + m) * P_;
  const int*   xib = xi + b * P_;
  for (int p = 0; p < P_; ++p) {
    const int t = xib[p];
    xsb[t * N_ + m] += xb[p];
  }
}

// ---------------------------------------------------------------------------
// Kernel 4: Zpart[kch] = Xs[:, kch] @ fT[kch, :] via V_WMMA_F32_16X16X4_F32.
// Workgroup = 8 waves; wave w owns C rows [16w,16w+16) of one 16-col stripe.
// B panel (shared by all 8 waves) is double-buffered in LDS via
// global_load_async_to_lds_b64 + s_wait_asynccnt (CDNA5 async path).
// K split across gridDim.y for wave parallelism; deterministic reduce after.
// ---------------------------------------------------------------------------
__global__ __launch_bounds__(256)
void z_wmma_kernel(const float* __restrict__ Xs,    // [B_, K_]
                   const float* __restrict__ fT,    // [K_, NPAD]
                   float* __restrict__ Zpart) {     // [KS_, B_, NPAD]
  __shared__ float bpanel[2][TK_ * 16];
  const int tileN = blockIdx.x;        // 0..6
  const int kch   = blockIdx.y;        // 0..KS_-1
  const int tid   = threadIdx.x;       // 0..255
  const int lane  = tid & 31;
  const int wave  = tid >> 5;          // 0..7
  const int half  = lane >> 4;         // 0 or 1
  const int l16   = lane & 15;
  const int kbase = kch * KC_;

  const int    rowA = wave * 16 + l16;
  const float* Arow = Xs + (size_t)rowA * K_ + kbase;

  // Each thread asynchronously copies one float2 of the B stage into LDS:
  // element pair e -> (row = e>>3, cols 2*(e&7)..+1) of the TK_ x 16 stage.
  const int    brow  = tid >> 3;                  // 0..31
  const int    bcol2 = (tid & 7) * 2;             // 0,2,..,14
  const float* gsrc0 = fT + (size_t)(kbase + brow) * NPAD + tileN * 16 + bcol2;

  {  // issue stage 0
    unsigned lds = (unsigned)(uintptr_t)&bpanel[0][brow * 16 + bcol2];
    asm volatile("global_load_async_to_lds_b64 %0, %1, off"
                 :: "v"(lds), "v"(gsrc0) : "memory");
  }

  v8f c = {};
  for (int s = 0; s < NSTG; ++s) {
    const int cur = s & 1;
    asm volatile("s_wait_asynccnt 0" ::: "memory");  // my stage-s copies done
    __syncthreads();                                 // everyone's are done
    if (s + 1 < NSTG) {                              // prefetch next stage
      unsigned lds = (unsigned)(uintptr_t)&bpanel[1 - cur][brow * 16 + bcol2];
      const float* gsrc = gsrc0 + (size_t)(s + 1) * TK_ * NPAD;
      asm volatile("global_load_async_to_lds_b64 %0, %1, off"
                   :: "v"(lds), "v"(gsrc) : "memory");
    }
    const float* As = Arow + s * TK_;
#pragma unroll
    for (int kk = 0; kk < TK_; kk += 4) {
      const int ka = kk + (half << 1);               // K=0,1 / K=2,3 halves
      const v2f a = *(const v2f*)(As + ka);          // 8B-aligned
      v2f b;
      b.x = bpanel[cur][ka * 16 + l16];              // ds_load, bank-clean
      b.y = bpanel[cur][(ka + 1) * 16 + l16];
      c = __builtin_amdgcn_wmma_f32_16x16x4_f32(false, a, false, b,
                                                (short)0, c, false, false);
    }
    __syncthreads();                                 // bpanel[cur] reusable
  }

  const int col   = tileN * 16 + l16;
  const int rbase = wave * 16 + half * 8;            // C/D VGPR r -> M=r(+8)
  float* zp = Zpart + (size_t)kch * B_ * NPAD;
#pragma unroll
  for (int r = 0; r < 8; ++r)
    zp[(size_t)(rbase + r) * NPAD + col] = c[r];
}

// ---------------------------------------------------------------------------
// Kernel 5: deterministic K-split reduction  Z[b,n] = sum_ks Zpart[ks,b,n].
// ---------------------------------------------------------------------------
__global__ __launch_bounds__(256)
void z_reduce_kernel(const float* __restrict__ Zpart,
                     float* __restrict__ Z) {
  const int idx = blockIdx.x * 256 + threadIdx.x;
  if (idx >= B_ * N_) return;
  const int b = idx / N_;
  const int n = idx - b * N_;
  float s = 0.f;
  for (int ks = 0; ks < KS_; ++ks)
    s += Zpart[((size_t)ks * B_ + b) * NPAD + n];
  Z[idx] = s;
}

// ---------------------------------------------------------------------------
extern "C" void kernel_launch(void* const* d_in, const int* in_sizes, int n_in,
                              void* d_out, int out_size, void* d_ws, size_t ws_size,
                              hipStream_t stream) {
  (void)in_sizes; (void)n_in; (void)out_size; (void)ws_size;
  const float* x      = (const float*)d_in[0];   // [B, N, P]
  const int*   xi     = (const int*)d_in[1];     // [B, P]
  const float* g      = (const float*)d_in[2];   // [N2, N2]
  const float* wts    = (const float*)d_in[3];   // [N2, 1]
  const float* alphas = (const float*)d_in[4];   // [T, 1]

  float* out = (float*)d_out;
  float* Z   = out;                 // [B, N]  = 12800 floats
  float* F   = out + B_ * N_;       // [N2, 1] = 10000 floats

  float* fT    = (float*)d_ws;                          // [K_, NPAD]
  float* Xs    = fT + (size_t)K_ * NPAD;                // [B_, K_]
  float* Zpart = Xs + (size_t)B_ * K_;                  // [KS_, B_, NPAD]
  const size_t zero_bytes =
      ((size_t)K_ * NPAD + (size_t)B_ * K_) * sizeof(float);
  hipMemsetAsync(d_ws, 0, zero_bytes, stream);    // zeros Xs + fT pad cols

  f_gemv_kernel<<<N2_, 256, 0, stream>>>(g, wts, F);
  softmax_kernel<<<K_, 128, 0, stream>>>(F, alphas, fT);
  xs_build_kernel<<<B_, 128, 0, stream>>>(x, xi, Xs);
  dim3 zgrid(NPAD / 16, KS_);                     // 7 x 25 workgroups, 8 waves
  z_wmma_kernel<<<zgrid, 256, 0, stream>>>(Xs, fT, Zpart);
  z_reduce_kernel<<<(B_ * N_ + 255) / 256, 256, 0, stream>>>(Zpart, Z);
}